// SoftplusRNN_81724637708789
// MI455X (gfx1250) — compile-verified
//
#include <hip/hip_runtime.h>
#include <math.h>

typedef float v2f __attribute__((ext_vector_type(2)));
typedef float v8f __attribute__((ext_vector_type(8)));

constexpr int BATCH = 32;
constexpr int T     = 8192;
constexpr int D     = 128;
constexpr int CH    = 64;    // chunks along T
constexpr int L     = 128;   // chunk length; CH*L == T; one chunk == one block tile

// ---- CDNA5 async global<->LDS path (guarded; falls back to float4 staging) ----
#if defined(__gfx1250__) && \
    __has_builtin(__builtin_amdgcn_global_load_async_to_lds_b128) && \
    __has_builtin(__builtin_amdgcn_global_store_async_from_lds_b128)
#define ASYNC_LDS 1
#else
#define ASYNC_LDS 0
#endif

typedef int v4i __attribute__((ext_vector_type(4)));
typedef __attribute__((address_space(1))) v4i gv4i;   // global int4 (AS1)
typedef __attribute__((address_space(3))) v4i lv4i;   // LDS int4 (AS3)

__device__ __forceinline__ void async_wait0() {
#if __has_builtin(__builtin_amdgcn_s_wait_asynccnt)
    __builtin_amdgcn_s_wait_asynccnt(0);
#else
    asm volatile("s_wait_asynccnt 0x0" ::: "memory");
#endif
}

// Stage n4 float4's from global g -> LDS l with 256 threads.
__device__ __forceinline__ void stage_tile(const float* g, float* l, int n4, int tid) {
#if ASYNC_LDS
    for (int i = tid; i < n4; i += 256) {
        __builtin_amdgcn_global_load_async_to_lds_b128(
            (gv4i*)((const float4*)g + i), (lv4i*)((float4*)l + i), 0, 0);
    }
#else
    for (int i = tid; i < n4; i += 256) {
        ((float4*)l)[i] = ((const float4*)g)[i];
    }
#endif
}

// Store n4 float4's from LDS l -> global g (async; ENDPGM drains ASYNCcnt).
__device__ __forceinline__ void store_tile(float* g, const float* l, int n4, int tid) {
#if ASYNC_LDS
    for (int i = tid; i < n4; i += 256) {
        __builtin_amdgcn_global_store_async_from_lds_b128(
            (gv4i*)((float4*)g + i), (lv4i*)((const float4*)l + i), 0, 0);
    }
#else
    for (int i = tid; i < n4; i += 256) {
        ((float4*)g)[i] = ((const float4*)l)[i];
    }
#endif
}

__device__ __forceinline__ float softplus_f(float x) {
    return fmaxf(x, 0.0f) + log1pf(expf(-fabsf(x)));
}

// Compute a 128x128 output tile: acc[nt] (nt = column tile 0..7) for this wave.
// 8 waves x 16 rows each; K = 128 in steps of 4 via V_WMMA_F32_16X16X4_F32.
// BT: B = P^T (B[K,n] = P[j,K]) else B = P (B[K,n] = P[K,j]).
template<bool BT>
__device__ __forceinline__ void gemm_tile_compute(const float* sA, const float* sP,
                                                  int tid, v8f acc[8]) {
    const int lane = tid & 31;
    const int wave = tid >> 5;        // 0..7
    const int m    = lane & 15;
    const int h    = lane >> 4;
    #pragma unroll
    for (int nt = 0; nt < 8; ++nt) acc[nt] = (v8f){};
    for (int k0 = 0; k0 < D; k0 += 4) {
        const int ka = k0 + 2 * h;
        v2f a;
        a.x = sA[(wave * 16 + m) * D + ka];
        a.y = sA[(wave * 16 + m) * D + ka + 1];
        #pragma unroll
        for (int nt = 0; nt < 8; ++nt) {
            const int j = nt * 16 + m;
            v2f b;
            if (BT) { b.x = sP[j * D + ka];  b.y = sP[j * D + ka + 1]; }
            else    { b.x = sP[ka * D + j];  b.y = sP[(ka + 1) * D + j]; }
            acc[nt] = __builtin_amdgcn_wmma_f32_16x16x4_f32(
                false, a, false, b, (short)0, acc[nt], false, false);
        }
    }
}

// Park this wave's 16x128 result strip into LDS tile (row-major L x D).
__device__ __forceinline__ void park_tile(float* sA, int tid, const v8f acc[8]) {
    const int lane = tid & 31, wave = tid >> 5, m = lane & 15, h = lane >> 4;
    #pragma unroll
    for (int nt = 0; nt < 8; ++nt) {
        #pragma unroll
        for (int r = 0; r < 8; ++r) {
            sA[(wave * 16 + r + 8 * h) * D + nt * 16 + m] = acc[nt][r];
        }
    }
}

// K1: Y-tile = X-tile @ P^T; park tile in LDS; async-store Y; chunk scan -> carry.
__global__ __launch_bounds__(256)
void gemm1_scan_kernel(const float* __restrict__ x, const float* __restrict__ P,
                       const float* __restrict__ w,
                       float* __restrict__ Y, float* __restrict__ carry) {
    __shared__ __align__(16) float sA[L * D];   // 64 KB: X tile -> Y tile
    __shared__ __align__(16) float sP[D * D];   // 64 KB
    const int  tid     = threadIdx.x;
    const long rowBase = (long)blockIdx.x * L;  // blockIdx = b*CH + c

    stage_tile(P, sP, (D * D) / 4, tid);
    stage_tile(x + rowBase * D, sA, (L * D) / 4, tid);
#if ASYNC_LDS
    async_wait0();
#endif
    __syncthreads();

    v8f acc[8];
    gemm_tile_compute<true>(sA, sP, tid, acc);
    __syncthreads();                 // all waves done reading sA before overwrite

    park_tile(sA, tid, acc);         // Y tile now in LDS
    __syncthreads();                 // LDS writes visible (waits DScnt)

    // coalesced async store of the Y tile; ENDPGM drains ASYNCcnt
    store_tile(Y + rowBase * D, sA, (L * D) / 4, tid);

    if (tid < D) {                   // zero-init scan of this chunk, one d per thread
        const int   d = tid;
        const float a = 1.0f - softplus_f(w[d]);
        float s = 0.0f;
        for (int i = 0; i < L; ++i) s = fmaf(a, s, sA[i * D + d]);
        carry[(long)blockIdx.x * D + d] = s;
    }
}

// K2: sequential combine of CH chunk carries per (b,d); carry[] becomes carry-IN.
__global__ __launch_bounds__(128)
void scan_combine_kernel(const float* __restrict__ w, float* carry) {
    const int idx = blockIdx.x * 128 + threadIdx.x;  // [0, BATCH*D)
    const int d = idx & (D - 1);
    const int b = idx >> 7;
    const float a = 1.0f - softplus_f(w[d]);
    float aL = a;                     // a^128 via 7 squarings (sign-exact)
    #pragma unroll
    for (int i = 0; i < 7; ++i) aL *= aL;
    float run = 0.0f;
    float* cp = carry + (long)b * CH * D + d;
    for (int c = 0; c < CH; ++c) {
        const float sc = cp[(long)c * D];
        cp[(long)c * D] = run;        // carry-in for chunk c
        run = fmaf(aL, run, sc);
    }
}

// K3: stage Y chunk -> seeded scan in LDS (Y -> U) -> H = U @ P -> async store H.
// In place on d_out: whole chunk staged to LDS before any global write.
__global__ __launch_bounds__(256)
void scan_gemm2_kernel(float* YH, const float* __restrict__ P,
                       const float* __restrict__ w, const float* __restrict__ carryIn) {
    __shared__ __align__(16) float sA[L * D];
    __shared__ __align__(16) float sP[D * D];
    const int  tid     = threadIdx.x;
    const long rowBase = (long)blockIdx.x * L;

    stage_tile(P, sP, (D * D) / 4, tid);
    stage_tile(YH + rowBase * D, sA, (L * D) / 4, tid);
#if ASYNC_LDS
    async_wait0();
#endif
    __syncthreads();

    if (tid < D) {                    // seeded scan, overwrite LDS tile with u
        const int   d = tid;
        const float a = 1.0f - softplus_f(w[d]);
        float s = carryIn[(long)blockIdx.x * D + d];
        for (int i = 0; i < L; ++i) {
            s = fmaf(a, s, sA[i * D + d]);
            sA[i * D + d] = s;
        }
    }
    __syncthreads();

    v8f acc[8];
    gemm_tile_compute<false>(sA, sP, tid, acc);
    __syncthreads();                  // everyone done reading U tile from sA

    park_tile(sA, tid, acc);          // H tile now in LDS
    __syncthreads();                  // LDS writes visible before async reads

    store_tile(YH + rowBase * D, sA, (L * D) / 4, tid);  // coalesced H store
}

extern "C" void kernel_launch(void* const* d_in, const int* in_sizes, int n_in,
                              void* d_out, int out_size, void* d_ws, size_t ws_size,
                              hipStream_t stream) {
    const float* x = (const float*)d_in[0];   // (B,T,D)
    const float* w = (const float*)d_in[1];   // (D,)
    const float* P = (const float*)d_in[2];   // (D,D)
    float* out   = (float*)d_out;             // reused: Y -> (in place) -> H
    float* carry = (float*)d_ws;              // BATCH*CH*D floats = 1 MB

    const int nChunks = BATCH * CH;           // 2048 blocks, one chunk each

    gemm1_scan_kernel<<<nChunks, 256, 0, stream>>>(x, P, w, out, carry);
    scan_combine_kernel<<<(BATCH * D) / 128, 128, 0, stream>>>(w, carry);
    scan_gemm2_kernel<<<nChunks, 256, 0, stream>>>(out, P, w, carry);
}